// SMPLXLayer_39805756899441
// MI455X (gfx1250) — compile-verified
//
#include <hip/hip_runtime.h>

#define BATCH 256
#define NV 10475
#define NJ 55
#define N3 (NV * 3)          // 31425
#define KPOSE 486            // (NJ-1)*9
#define KP 512               // padded K for WMMA
#define NBETA 10

typedef __attribute__((ext_vector_type(2))) float v2f;
typedef __attribute__((ext_vector_type(8))) float v8f;

__constant__ int c_parents[NJ] = {
    -1, 0, 0, 0, 1, 2, 3, 4, 5, 6, 7, 8, 9, 9, 9, 12, 13, 14, 16, 17, 18,
    19, 15, 15, 15, 20, 25, 26, 20, 28, 29, 20, 31, 32, 20, 34, 35, 20, 37,
    38, 21, 40, 41, 21, 43, 44, 21, 46, 47, 21, 49, 50, 21, 52, 53};

// ---------------------------------------------------------------- full_pose / pose_feature
__device__ __forceinline__ float fetch_pose(int b, int j, int rc,
    const float* wr, const float* bp, const float* jaw, const float* le,
    const float* re, const float* lh, const float* rh)
{
    if (j == 0)  return wr[b * 9 + rc];
    if (j <= 21) return bp[(b * 21 + (j - 1)) * 9 + rc];
    if (j == 22) return jaw[b * 9 + rc];
    if (j == 23) return le[b * 9 + rc];
    if (j == 24) return re[b * 9 + rc];
    if (j <= 39) return lh[(b * 15 + (j - 25)) * 9 + rc];
    return rh[(b * 15 + (j - 40)) * 9 + rc];
}

__global__ void pack_pose_kernel(const float* __restrict__ wr, const float* __restrict__ bp,
                                 const float* __restrict__ jaw, const float* __restrict__ le,
                                 const float* __restrict__ re, const float* __restrict__ lh,
                                 const float* __restrict__ rh,
                                 float* __restrict__ fullpose, float* __restrict__ pf)
{
    const int b = blockIdx.x;
    const int t = threadIdx.x;               // 512 threads
    if (t < NJ * 9)
        fullpose[b * NJ * 9 + t] = fetch_pose(b, t / 9, t % 9, wr, bp, jaw, le, re, lh, rh);
    float v = 0.0f;
    if (t < KPOSE) {
        const int j = t / 9 + 1, rc = t % 9;
        const float diag = (rc == 0 || rc == 4 || rc == 8) ? 1.0f : 0.0f;
        v = fetch_pose(b, j, rc, wr, bp, jaw, le, re, lh, rh) - diag;
    }
    pf[b * KP + t] = v;                      // zero pad K in [486, 512)
}

// ---------------------------------------------------------------- v_shaped
__global__ void vshaped_kernel(const float* __restrict__ vt, const float* __restrict__ sd,
                               const float* __restrict__ ed, const float* __restrict__ bsh,
                               const float* __restrict__ esh, float* __restrict__ vsh)
{
    const int n = blockIdx.x * blockDim.x + threadIdx.x;
    const int b = blockIdx.y;
    if (n >= NV) return;
    float a0 = vt[n * 3 + 0], a1 = vt[n * 3 + 1], a2 = vt[n * 3 + 2];
    const float* bs = bsh + b * NBETA;
    const float* es = esh + b * NBETA;
    const float* s  = sd + (size_t)n * 30;
    const float* e  = ed + (size_t)n * 30;
#pragma unroll
    for (int k = 0; k < NBETA; ++k) {
        const float wb = bs[k], we = es[k];
        a0 += s[k]      * wb + e[k]      * we;
        a1 += s[10 + k] * wb + e[10 + k] * we;
        a2 += s[20 + k] * wb + e[20 + k] * we;
    }
    float* o = vsh + (size_t)b * N3 + n * 3;
    o[0] = a0; o[1] = a1; o[2] = a2;
}

// ---------------------------------------------------------------- joints = Jreg @ v_shaped
__global__ void joints_kernel(const float* __restrict__ jreg, const float* __restrict__ vsh,
                              float* __restrict__ joints)
{
    const int j = blockIdx.x;                // 55
    const int b = blockIdx.y;                // 256
    const int t = threadIdx.x;               // 64
    float s0 = 0.f, s1 = 0.f, s2 = 0.f;
    const float* row = jreg + (size_t)j * NV;
    const float* v   = vsh + (size_t)b * N3;
    for (int n = t; n < NV; n += 64) {
        const float w = row[n];
        s0 += w * v[n * 3 + 0];
        s1 += w * v[n * 3 + 1];
        s2 += w * v[n * 3 + 2];
    }
    __shared__ float red[64 * 3];
    red[t * 3 + 0] = s0; red[t * 3 + 1] = s1; red[t * 3 + 2] = s2;
    __syncthreads();
    if (t < 3) {
        float acc = 0.f;
        for (int i = 0; i < 64; ++i) acc += red[i * 3 + t];
        joints[(b * NJ + j) * 3 + t] = acc;
    }
}

// ---------------------------------------------------------------- kinematic chain -> A_rel
__global__ void chain_kernel(const float* __restrict__ fullpose, const float* __restrict__ joints,
                             const float* __restrict__ wtsl, float* __restrict__ Amat,
                             float* __restrict__ jout)
{
    const int b = blockIdx.x * blockDim.x + threadIdx.x;
    if (b >= BATCH) return;
    const float* fp = fullpose + b * NJ * 9;
    const float* J  = joints + b * NJ * 3;
    float* A  = Amat + b * NJ * 12;
    float* jo = jout + b * NJ * 3;
    const float wx = wtsl[b * 3 + 0], wy = wtsl[b * 3 + 1], wz = wtsl[b * 3 + 2];
    for (int j = 0; j < NJ; ++j) {
        const int p = c_parents[j];
        float rj0 = J[j * 3 + 0], rj1 = J[j * 3 + 1], rj2 = J[j * 3 + 2];
        if (p >= 0) { rj0 -= J[p * 3 + 0]; rj1 -= J[p * 3 + 1]; rj2 -= J[p * 3 + 2]; }
        const float* R = fp + j * 9;
        float o[12];
        if (p < 0) {
            o[0] = R[0]; o[1] = R[1]; o[2] = R[2];  o[3]  = rj0;
            o[4] = R[3]; o[5] = R[4]; o[6] = R[5];  o[7]  = rj1;
            o[8] = R[6]; o[9] = R[7]; o[10] = R[8]; o[11] = rj2;
        } else {
            const float* P = A + p * 12;          // already written by this thread
#pragma unroll
            for (int r = 0; r < 3; ++r) {
                const float p0 = P[r * 4 + 0], p1 = P[r * 4 + 1],
                            p2 = P[r * 4 + 2], p3 = P[r * 4 + 3];
                o[r * 4 + 0] = p0 * R[0] + p1 * R[3] + p2 * R[6];
                o[r * 4 + 1] = p0 * R[1] + p1 * R[4] + p2 * R[7];
                o[r * 4 + 2] = p0 * R[2] + p1 * R[5] + p2 * R[8];
                o[r * 4 + 3] = p0 * rj0 + p1 * rj1 + p2 * rj2 + p3;
            }
        }
#pragma unroll
        for (int i = 0; i < 12; ++i) A[j * 12 + i] = o[i];
        jo[j * 3 + 0] = o[3] + wx;
        jo[j * 3 + 1] = o[7] + wy;
        jo[j * 3 + 2] = o[11] + wz;
    }
    // A_rel: t -= R @ joints[j]
    for (int j = 0; j < NJ; ++j) {
        const float jx = J[j * 3], jy = J[j * 3 + 1], jz = J[j * 3 + 2];
        float* Aj = A + j * 12;
        Aj[3]  -= Aj[0] * jx + Aj[1] * jy + Aj[2]  * jz;
        Aj[7]  -= Aj[4] * jx + Aj[5] * jy + Aj[6]  * jz;
        Aj[11] -= Aj[8] * jx + Aj[9] * jy + Aj[10] * jz;
    }
}

// ---------------------------------------------------------------- lbs_weights transpose
__global__ void transpose_w_kernel(const float* __restrict__ w, float* __restrict__ wT)
{
    const int n = blockIdx.x * blockDim.x + threadIdx.x;
    const int j = blockIdx.y;
    if (n < NV) wT[(size_t)j * NV + n] = w[(size_t)n * NJ + j];
}

// ---------------------------------------------------------------- WMMA GEMM: vsh += pf @ posedirs
// M=256 (batch), K=486 (padded 512 in pf), N=31425. Wave tile: 32(M) x 64(N).
// Branch-free inner loop: N tail handled by clamping lane columns (bad columns
// never stored), K tail handled by clamping the B row (A rows 486/487 are zero
// in pf, so the clamped-B product contributes exactly zero).
__global__ void __launch_bounds__(256) pose_gemm_kernel(const float* __restrict__ pf,
                                                        const float* __restrict__ pd,
                                                        float* __restrict__ vsh)
{
    const int lane = threadIdx.x & 31;
    const int wave = threadIdx.x >> 5;
    const int NTN  = (N3 + 63) >> 6;              // 492
    const int widx = blockIdx.x * 8 + wave;
    const int mt   = widx / NTN;
    const int nt   = widx - mt * NTN;
    if (mt >= BATCH / 32) return;
    const int m0  = mt << 5;                      // 32 rows of M per wave
    const int n0  = nt << 6;                      // 64 cols of N per wave
    const int lh  = lane >> 4;                    // 0/1: lane half
    const int l16 = lane & 15;
    const int kA  = lh * 2;                       // this lane's K pair within 4

    // clamped (always-valid) column index per N subtile
    int nA[4];
#pragma unroll
    for (int t = 0; t < 4; ++t) {
        int n = n0 + t * 16 + l16;
        nA[t] = n < N3 ? n : (N3 - 1);
    }

    v8f acc[8];
#pragma unroll
    for (int i = 0; i < 8; ++i) acc[i] = (v8f){};

    const float* arow0 = pf + (size_t)(m0 + l16) * KP;       // M subtile 0
    const float* arow1 = arow0 + (size_t)16 * KP;            // M subtile 1

    for (int kb = 0; kb < 488; kb += 4) {
        const int k0  = kb + kA;                  // lane's first K (may be 486)
        const int k0B = k0 < (KPOSE - 2) ? k0 : (KPOSE - 2);  // clamp B row
        v2f a0, a1;
        a0.x = arow0[k0];
        a0.y = arow0[k0 + 1];
        a1.x = arow1[k0];
        a1.y = arow1[k0 + 1];
        const float* b0 = pd + (size_t)k0B * N3;
        const float* b1 = b0 + N3;
        if (kb + 4 < KPOSE)
            __builtin_prefetch(pd + (size_t)(kb + 4 + kA) * N3 + nA[0], 0, 1);
#pragma unroll
        for (int t = 0; t < 4; ++t) {
            v2f bb;
            bb.x = b0[nA[t]];
            bb.y = b1[nA[t]];
            acc[t]     = __builtin_amdgcn_wmma_f32_16x16x4_f32(false, a0, false, bb,
                                                               (short)0, acc[t], false, false);
            acc[4 + t] = __builtin_amdgcn_wmma_f32_16x16x4_f32(false, a1, false, bb,
                                                               (short)0, acc[4 + t], false, false);
        }
    }

    // D layout: VGPR r, lanes 0-15 -> M=r, lanes 16-31 -> M=r+8; N=lane&15
#pragma unroll
    for (int r = 0; r < 8; ++r) {
        const int m = m0 + r + lh * 8;
        float* out0 = vsh + (size_t)m * N3;
        float* out1 = vsh + (size_t)(m + 16) * N3;
#pragma unroll
        for (int t = 0; t < 4; ++t) {
            const int n = n0 + t * 16 + l16;
            if (n < N3) {
                out0[n] += acc[t][r];
                out1[n] += acc[4 + t][r];
            }
        }
    }
}

// ---------------------------------------------------------------- skinning
__global__ void skin_kernel(const float* __restrict__ vsh, const float* __restrict__ wT,
                            const float* __restrict__ Amat, const float* __restrict__ wtsl,
                            float* __restrict__ verts)
{
    const int b = blockIdx.y;
    const int n = blockIdx.x * blockDim.x + threadIdx.x;
    __shared__ float sA[NJ * 12];
    for (int i = threadIdx.x; i < NJ * 12; i += blockDim.x)
        sA[i] = Amat[(size_t)b * NJ * 12 + i];
    __syncthreads();
    if (n >= NV) return;
    const float* v = vsh + (size_t)b * N3 + n * 3;
    const float vx = v[0], vy = v[1], vz = v[2];
    float m[12];
#pragma unroll
    for (int i = 0; i < 12; ++i) m[i] = 0.f;
    for (int j = 0; j < NJ; ++j) {
        const float w = wT[(size_t)j * NV + n];
        const float* Aj = sA + j * 12;
#pragma unroll
        for (int i = 0; i < 12; ++i) m[i] += w * Aj[i];
    }
    const float ox = m[0] * vx + m[1] * vy + m[2]  * vz + m[3]  + wtsl[b * 3 + 0];
    const float oy = m[4] * vx + m[5] * vy + m[6]  * vz + m[7]  + wtsl[b * 3 + 1];
    const float oz = m[8] * vx + m[9] * vy + m[10] * vz + m[11] + wtsl[b * 3 + 2];
    float* o = verts + (size_t)b * N3 + n * 3;
    o[0] = ox; o[1] = oy; o[2] = oz;
}

// ---------------------------------------------------------------- launch
extern "C" void kernel_launch(void* const* d_in, const int* in_sizes, int n_in,
                              void* d_out, int out_size, void* d_ws, size_t ws_size,
                              hipStream_t stream)
{
    const float* wr  = (const float*)d_in[0];   // world_rot    (B,1,3,3)
    const float* wt  = (const float*)d_in[1];   // world_tsl    (B,3)
    const float* bsh = (const float*)d_in[2];   // body_shape   (B,10)
    const float* bp  = (const float*)d_in[3];   // body_pose    (B,21,3,3)
    const float* lh  = (const float*)d_in[4];   // left_hand    (B,15,3,3)
    const float* rh  = (const float*)d_in[5];   // right_hand   (B,15,3,3)
    const float* esh = (const float*)d_in[6];   // expr_shape   (B,10)
    const float* jaw = (const float*)d_in[7];   // jaw_pose     (B,1,3,3)
    const float* le  = (const float*)d_in[8];   // leye_pose
    const float* re  = (const float*)d_in[9];   // reye_pose
    const float* vt  = (const float*)d_in[10];  // v_template   (NV,3)
    const float* sd  = (const float*)d_in[11];  // shapedirs    (NV,3,10)
    const float* ed  = (const float*)d_in[12];  // expr_dirs    (NV,3,10)
    const float* pd  = (const float*)d_in[13];  // posedirs     (486, NV*3)
    const float* jr  = (const float*)d_in[14];  // J_regressor  (55, NV)
    const float* lw  = (const float*)d_in[15];  // lbs_weights  (NV, 55)

    float* out = (float*)d_out;
    float* ws  = (float*)d_ws;

    float* vsh      = ws;                                   // B*N3 (v_shaped -> v_posed in place)
    float* pf       = vsh + (size_t)BATCH * N3;             // B*KP  (zero-padded pose_feature)
    float* fullpose = pf + (size_t)BATCH * KP;              // B*55*9
    float* joints   = fullpose + (size_t)BATCH * NJ * 9;    // B*55*3
    float* Amat     = joints + (size_t)BATCH * NJ * 3;      // B*55*12 (A_rel)
    float* wT       = Amat + (size_t)BATCH * NJ * 12;       // 55*NV

    float* verts = out;                                     // B*NV*3
    float* jout  = out + (size_t)BATCH * N3;                // B*55*3

    pack_pose_kernel<<<BATCH, 512, 0, stream>>>(wr, bp, jaw, le, re, lh, rh, fullpose, pf);
    vshaped_kernel<<<dim3((NV + 255) / 256, BATCH), 256, 0, stream>>>(vt, sd, ed, bsh, esh, vsh);
    joints_kernel<<<dim3(NJ, BATCH), 64, 0, stream>>>(jr, vsh, joints);
    chain_kernel<<<(BATCH + 63) / 64, 64, 0, stream>>>(fullpose, joints, wt, Amat, jout);
    transpose_w_kernel<<<dim3((NV + 255) / 256, NJ), 256, 0, stream>>>(lw, wT);

    const int ntn = (N3 + 63) / 64;                          // 492 N tiles
    const int nwaves = (BATCH / 32) * ntn;                   // 3936 waves
    pose_gemm_kernel<<<(nwaves + 7) / 8, 256, 0, stream>>>(pf, pd, vsh);

    skin_kernel<<<dim3((NV + 255) / 256, BATCH), 256, 0, stream>>>(vsh, wT, Amat, wt, verts);
}